// MLC_31645319037046
// MI455X (gfx1250) — compile-verified
//
#include <hip/hip_runtime.h>
#include <hip/hip_bf16.h>

typedef float v2f __attribute__((ext_vector_type(2)));
typedef float v8f __attribute__((ext_vector_type(8)));
typedef int   v4i __attribute__((ext_vector_type(4)));
typedef __attribute__((address_space(1))) v4i gv4i;   // global int4
typedef __attribute__((address_space(3))) v4i sv4i;   // LDS int4

#define D_IN    2048
#define NCLS    210
#define NTILES  14          // ceil(210/16)
#define NPAD    (NTILES*16) // 224
#define KSLAB   64
#define WSTRIDE 68          // 64 + 4 pad: bank = (4*row + col) % 64 -> conflict-free b64
#define NSLABS  (D_IN / KSLAB)              // 32
#define LDSVEC  ((NPAD * KSLAB / 4) / 256)  // 14 float4 copies per thread per slab
#define K_TOP   10
#define D_SEM   512

// --- CDNA5 async global->LDS copy (no VGPR staging, tracked by ASYNCcnt) ----
__device__ __forceinline__ void async_copy_b128(const float* g, float* l) {
#if __has_builtin(__builtin_amdgcn_global_load_async_to_lds_b128)
  __builtin_amdgcn_global_load_async_to_lds_b128((gv4i*)g, (sv4i*)l, 0, 0);
#else
  const unsigned lds_off = (unsigned)(unsigned long long)(sv4i*)l;
  asm volatile("global_load_async_to_lds_b128 %0, %1, off"
               :: "v"(lds_off), "v"(g) : "memory");
#endif
}

__device__ __forceinline__ void wait_async0() {
#if __has_builtin(__builtin_amdgcn_s_wait_asynccnt)
  __builtin_amdgcn_s_wait_asynccnt(0);
#else
  asm volatile("s_wait_asynccnt 0x0" ::: "memory");
#endif
}

// ---------------------------------------------------------------------------
// tags[B,210] = A[B,2048] @ W[210,2048]^T + b   via V_WMMA_F32_16X16X4_F32
//
// Block = 256 threads = 8 waves; each wave owns 16 M-rows x ALL 14 N-tiles so
// every A element is fetched from global exactly once. W is staged in LDS in
// K-slabs of 64, double-buffered (2 x 59.5 KB), filled by ASYNC global->LDS
// b128 loads issued right after the barrier so they drain behind the 224 WMMAs
// of the current slab. One barrier per slab:
//   wait_asynccnt(0)  -> own slab-s loads landed
//   __syncthreads()   -> everyone's slab-s landed; idle buffer free (its last
//                        readers were in iteration s-1, before their barrier)
//   issue slab s+1    -> async engine fills idle buffer during compute
//   compute slab s
// ---------------------------------------------------------------------------
__global__ __launch_bounds__(256, 1)
void MLC_gemm_tags(const float* __restrict__ A, const float* __restrict__ W,
                   const float* __restrict__ bias, float* __restrict__ tags,
                   int B_rows) {
  __shared__ float wlds[2][NPAD * WSTRIDE];   // 2 * 60928 B = 121856 B

  const int lane = threadIdx.x & 31;
  const int wave = threadIdx.x >> 5;
  const int half = lane >> 4;              // selects K pair {0,1} vs {2,3}
  const int sub  = lane & 15;

  const int tileM = blockIdx.x * 8 + wave;
  const int m  = tileM * 16 + sub;
  const int mc = m < B_rows ? m : (B_rows - 1);
  const float* __restrict__ arow = A + (size_t)mc * D_IN + 2 * half;

  // Per-thread slab-copy geometry (slab-invariant).
  int srcOff[LDSVEC];   // float offset into W for kb=0
  int dstOff[LDSVEC];   // float offset into an LDS buffer (16B aligned)
  #pragma unroll
  for (int i = 0; i < LDSVEC; ++i) {
    const int flat = threadIdx.x + 256 * i;
    const int row  = flat >> 4;                 // 0..223
    const int col  = (flat & 15) << 2;          // 0..60
    const int rc   = row < NCLS ? row : (NCLS - 1);
    srcOff[i] = rc * D_IN + col;
    dstOff[i] = row * WSTRIDE + col;
  }

  v8f acc[NTILES];
  #pragma unroll
  for (int t = 0; t < NTILES; ++t) acc[t] = (v8f){};

  // Prologue: slab 0 -> buf 0 (async).
  #pragma unroll
  for (int i = 0; i < LDSVEC; ++i)
    async_copy_b128(W + srcOff[i], &wlds[0][dstOff[i]]);

  for (int s = 0; s < NSLABS; ++s) {
    wait_async0();        // own slab-s copies complete
    __syncthreads();      // all waves' slab-s visible; idle buffer reusable

    if (s + 1 < NSLABS) {
      float* __restrict__ nbuf = wlds[(s + 1) & 1];
      const int kn = (s + 1) * KSLAB;
      #pragma unroll
      for (int i = 0; i < LDSVEC; ++i)
        async_copy_b128(W + srcOff[i] + kn, &nbuf[dstOff[i]]);
    }

    const int kb = s * KSLAB;
    const float* __restrict__ buf = wlds[s & 1];
    #pragma unroll 2
    for (int kk = 0; kk < KSLAB; kk += 4) {
      const v2f a = *(const v2f*)(arow + kb + kk);
      #pragma unroll
      for (int t = 0; t < NTILES; ++t) {
        const v2f b = *(const v2f*)(&buf[(t * 16 + sub) * WSTRIDE + kk + 2 * half]);
        acc[t] = __builtin_amdgcn_wmma_f32_16x16x4_f32(
            false, a, false, b, (short)0, acc[t], false, false);
      }
    }
  }

  // Epilogue: bias add + store. Tile-uniform bounds path keeps it branch-light.
  const int  mbase = tileM * 16 + 8 * half;
  const bool mfull = (tileM * 16 + 16) <= B_rows;
  #pragma unroll
  for (int t = 0; t < NTILES; ++t) {
    const int n = t * 16 + sub;
    if (n < NCLS) {
      const float bv = bias[n];
      if (mfull) {
        #pragma unroll
        for (int r = 0; r < 8; ++r)
          tags[(size_t)(mbase + r) * NCLS + n] = acc[t][r] + bv;
      } else {
        #pragma unroll
        for (int r = 0; r < 8; ++r)
          if (mbase + r < B_rows)
            tags[(size_t)(mbase + r) * NCLS + n] = acc[t][r] + bv;
      }
    }
  }
}

// ---------------------------------------------------------------------------
// Fused top-k (k=10, JAX tie-break: larger value, then smaller index) + gather.
// One wave per row of tags. Gather writes are lane-coalesced float4 (512B/inst).
// ---------------------------------------------------------------------------
__global__ __launch_bounds__(256)
void MLC_topk_gather(const float* __restrict__ tags,
                     const float* __restrict__ embed,
                     float* __restrict__ sem, int B_rows) {
  const int row  = blockIdx.x * (blockDim.x >> 5) + (threadIdx.x >> 5);
  const int lane = threadIdx.x & 31;
  if (row >= B_rows) return;

  const float* __restrict__ trow = tags + (size_t)row * NCLS;

  float vals[7];
  #pragma unroll
  for (int i = 0; i < 7; ++i) {
    const int c = lane + 32 * i;
    vals[i] = (c < NCLS) ? trow[c] : -__builtin_inff();
  }

  int top[K_TOP];
  #pragma unroll
  for (int j = 0; j < K_TOP; ++j) {
    float bv = vals[0];
    int   bs = 0;
    #pragma unroll
    for (int i = 1; i < 7; ++i) {
      if (vals[i] > bv) { bv = vals[i]; bs = i; }
    }
    int bidx = lane + 32 * bs;
    #pragma unroll
    for (int off = 16; off > 0; off >>= 1) {
      const float ov = __shfl_xor(bv,   off, 32);
      const int   oi = __shfl_xor(bidx, off, 32);
      if (ov > bv || (ov == bv && oi < bidx)) { bv = ov; bidx = oi; }
    }
    top[j] = bidx;
    if ((bidx & 31) == lane) vals[bidx >> 5] = -__builtin_inff();
  }

  float* __restrict__ orow = sem + (size_t)row * (K_TOP * D_SEM);
  #pragma unroll
  for (int j = 0; j < K_TOP; ++j) {
    const float4* __restrict__ src = (const float4*)(embed + (size_t)top[j] * D_SEM);
    float4* __restrict__ dst = (float4*)(orow + j * D_SEM);
    #pragma unroll
    for (int q = 0; q < 4; ++q) {
      dst[lane + 32 * q] = src[lane + 32 * q];
    }
  }
}

extern "C" void kernel_launch(void* const* d_in, const int* in_sizes, int n_in,
                              void* d_out, int out_size, void* d_ws, size_t ws_size,
                              hipStream_t stream) {
  const float* A     = (const float*)d_in[0];   // [B, 2048]
  const float* W     = (const float*)d_in[1];   // [210, 2048]
  const float* bias  = (const float*)d_in[2];   // [210]
  const float* embed = (const float*)d_in[3];   // [210, 512]
  const int B_rows   = in_sizes[0] / D_IN;      // 32768

  float* tags = (float*)d_out;
  float* sem  = (float*)d_out + (size_t)B_rows * NCLS;

  // GEMM: block = 128 M-rows x all 210 cols; 256 blocks for B=32768
  const int mBlocks = (B_rows + 127) / 128;
  MLC_gemm_tags<<<mBlocks, dim3(256), 0, stream>>>(A, W, bias, tags, B_rows);

  // top-k + gather: one wave per row, 8 waves per block
  const int nBlocks = (B_rows + 7) / 8;
  MLC_topk_gather<<<nBlocks, 256, 0, stream>>>(tags, embed, sem, B_rows);
}